// LGN_17042430230896
// MI455X (gfx1250) — compile-verified
//
#include <hip/hip_runtime.h>

// LGN gather + 7x7 conv == skinny GEMM: C[802816,64] = gather(A)[.,147] * W^T[147,64]
// fp32 via V_WMMA_F32_16X16X4_F32. Memory bound: 205 MB stores + ~470 MB L2 gather
// traffic; so (a) hoist all gather-address math out of the K loop into an LDS
// offset table, (b) coalesce stores with b128, (c) stage weights via async-to-LDS.

#define KSZ   7
#define IMG   224
#define NC    112
#define CIN   3
#define COUT  64
#define PADW  12                 // (K/2) * max dilation (4)
#define NPTS  (NC * NC)          // 12544
#define KTOT  (CIN * KSZ * KSZ)  // 147
#define KPAD  150                // even stride, zero-padded K
#define MBLK  64                 // M rows per block (divides NPTS)
#define TSTR  68                 // transpose buffer stride (16B-aligned float4)

typedef __attribute__((ext_vector_type(2))) float v2f;
typedef __attribute__((ext_vector_type(8))) float v8f;

typedef __attribute__((address_space(1))) int* gas_i;   // global (as1) int*
typedef __attribute__((address_space(3))) int* las_i;   // LDS (as3) int*

#ifdef __has_builtin
# if __has_builtin(__builtin_amdgcn_global_load_async_to_lds_b32)
#  define HAS_ASYNC_LDS 1
# endif
# if __has_builtin(__builtin_amdgcn_s_wait_asynccnt)
#  define HAS_WAIT_ASYNC 1
# endif
#endif
#ifndef HAS_ASYNC_LDS
# define HAS_ASYNC_LDS 0
#endif
#ifndef HAS_WAIT_ASYNC
# define HAS_WAIT_ASYNC 0
#endif

// ---------------------------------------------------------------------------
// Setup: replicate the numpy geometry (ci, cj, dil per grid point) into d_ws.
// ---------------------------------------------------------------------------
__global__ __launch_bounds__(256) void lgn_geometry_kernel(int* __restrict__ geo)
{
    __shared__ float redbuf[256];
    __shared__ float s_max;
    const int tid = threadIdx.x;

    float lmax = -1e30f;
    for (int p = tid; p < NPTS; p += 256) {
        const int i = p / NC, j = p % NC;
        const float gi = -1.0f + 2.0f * (float)i / (float)(NC - 1);
        const float gj = -1.0f + 2.0f * (float)j / (float)(NC - 1);
        const float ang = atan2f(gi, gj);
        const float ecc = sqrtf(gi * gi + gj * gj);
        const float le  = expf(ecc) - 1.0f;
        lmax = fmaxf(lmax, fmaxf(le * cosf(ang), le * sinf(ang)));
    }
    redbuf[tid] = lmax;
    __syncthreads();
    for (int s = 128; s > 0; s >>= 1) {
        if (tid < s) redbuf[tid] = fmaxf(redbuf[tid], redbuf[tid + s]);
        __syncthreads();
    }
    if (tid == 0) s_max = redbuf[0];
    __syncthreads();

    const float scale = ((float)(IMG / 2) - 0.5f) / s_max;  // 111.5 / max
    for (int p = tid; p < NPTS; p += 256) {
        const int i = p / NC, j = p % NC;
        const float gi = -1.0f + 2.0f * (float)i / (float)(NC - 1);
        const float gj = -1.0f + 2.0f * (float)j / (float)(NC - 1);
        const float ang = atan2f(gi, gj);
        const float ecc = sqrtf(gi * gi + gj * gj);
        const float le  = expf(ecc) - 1.0f;
        geo[p]            = (int)(le * cosf(ang) * scale + (float)(IMG / 2));
        geo[NPTS + p]     = (int)(le * sinf(ang) * scale + (float)(IMG / 2));
        geo[2 * NPTS + p] = (int)rintf(le) + 1;
    }
}

// ---------------------------------------------------------------------------
// Main: block -> 64(M) x 64(N); 8 wave32s = 4 M-tiles x 2 N-halves,
// each wave holds two 16x16 f32 accumulators sharing one A fragment.
// ---------------------------------------------------------------------------
__global__ __launch_bounds__(256) void lgn_wmma_kernel(
    const float* __restrict__ in,   // [64,3,224,224]
    const float* __restrict__ w,    // [64,147]
    const int*   __restrict__ geo,  // ci | cj | dil (NPTS each)
    float*       __restrict__ out)  // [64,64,112,112]
{
    __shared__ __align__(16) float Wlds[COUT * KPAD];  // W^T rows, zero-padded K
    __shared__ __align__(16) int   offs[MBLK * KPAD];  // per-row gather offsets
    __shared__ __align__(16) float T[COUT * TSTR];     // store-transpose buffer
    __shared__ int rg[3 * MBLK];                       // per-row ci/cj/dil

    const int tid  = threadIdx.x;
    const int lane = tid & 31;
    const int wv   = tid >> 5;   // 0..7
    const int wm   = wv >> 1;    // 0..3 : M sub-tile
    const int wn   = wv & 1;     // 0..1 : N half (2 x 16 cols)
    const int l15  = lane & 15;
    const int kh   = lane >> 4;  // K-half of the 16x16x4 operand layout

    const int mblk = blockIdx.x * MBLK;
    const int bb   = mblk / NPTS;          // constant per block (64 | NPTS)
    const int ij0  = mblk - bb * NPTS;

    // ---- zero-pad weight tail k in [147, KPAD) ---------------------------
    for (int idx = tid; idx < COUT * (KPAD - KTOT); idx += 256) {
        const int o = idx / (KPAD - KTOT);
        const int k = KTOT + idx - o * (KPAD - KTOT);
        Wlds[o * KPAD + k] = 0.0f;
    }
    // ---- stage weights (async-to-LDS when available) ---------------------
#if HAS_ASYNC_LDS
    for (int idx = tid; idx < COUT * KTOT; idx += 256) {
        const int o = idx / KTOT, k = idx - o * KTOT;
        __builtin_amdgcn_global_load_async_to_lds_b32(
            (gas_i)(int*)(w + idx), (las_i)(int*)&Wlds[o * KPAD + k], 0, 0);
    }
#else
    if (tid == 0) __builtin_prefetch(w, 0, 1);
    for (int idx = tid; idx < COUT * KTOT; idx += 256) {
        const int o = idx / KTOT, k = idx - o * KTOT;
        Wlds[o * KPAD + k] = w[idx];
    }
#endif
    // ---- per-row geometry ------------------------------------------------
    if (tid < MBLK) {
        const int ij = ij0 + tid;
        rg[tid]            = geo[ij];
        rg[MBLK + tid]     = geo[NPTS + ij];
        rg[2 * MBLK + tid] = geo[2 * NPTS + ij];
    }
    __syncthreads();

    // ---- build per-row gather-offset table (all address math lives here) --
    {
        const int r   = tid & 63;             // this thread's row
        const int ci  = rg[r];
        const int cj  = rg[MBLK + r];
        const int dil = rg[2 * MBLK + r];
        const int cim = ci - PADW - 3 * dil;
        const int cjm = cj - PADW - 3 * dil;
        for (int e = (tid >> 6); e < 148; e += 4) {   // 37 entries/thread
            int so = (int)0x80000000;
            if (e < KTOT) {
                const int c   = e / 49;
                const int rem = e - c * 49;
                const int ka  = rem / 7;
                const int kb  = rem - ka * 7;
                const int qi  = cim + dil * ka;
                const int qj  = cjm + dil * kb;
                const bool ok = ((unsigned)qi < (unsigned)IMG) &
                                ((unsigned)qj < (unsigned)IMG);
                const int qic = qi < 0 ? 0 : (qi > IMG - 1 ? IMG - 1 : qi);
                const int qjc = qj < 0 ? 0 : (qj > IMG - 1 ? IMG - 1 : qj);
                const int off = (c * IMG + qic) * IMG + qjc;
                so = ok ? off : (off | (int)0x80000000);
            }
            offs[r * KPAD + e] = so;
        }
    }
#if HAS_ASYNC_LDS
# if HAS_WAIT_ASYNC
    __builtin_amdgcn_s_wait_asynccnt(0);
# else
    asm volatile("s_wait_asynccnt 0x0" ::: "memory");
# endif
#endif
    __syncthreads();

    // ---- K loop: 37 steps, 2 x V_WMMA_F32_16X16X4_F32 per step -----------
    const float* inb = in + (size_t)bb * CIN * IMG * IMG;   // uniform base
    const int obase = (wm * 16 + l15) * KPAD;               // A row offsets
    const int nb0   = (wn * 32 + l15) * KPAD;               // B col 0..15
    const int nb1   = nb0 + 16 * KPAD;                      // B col 16..31

    v8f acc0 = {}, acc1 = {};
    #pragma unroll 4
    for (int k0 = 0; k0 < 148; k0 += 4) {
        const int e0 = k0 + 2 * kh;                          // even -> b64 ok
        const int2 sp = *(const int2*)&offs[obase + e0];     // ds_load_b64
        const float v0 = inb[sp.x & 0x7fffffff];             // global_load_b32
        const float v1 = inb[sp.y & 0x7fffffff];             // global_load_b32
        v2f a;
        a.x = sp.x >= 0 ? v0 : 0.5f;                         // pad value 0.5
        a.y = sp.y >= 0 ? v1 : 0.5f;
        const v2f bf0 = *(const v2f*)&Wlds[nb0 + e0];        // ds_load_b64
        const v2f bf1 = *(const v2f*)&Wlds[nb1 + e0];        // ds_load_b64
        acc0 = __builtin_amdgcn_wmma_f32_16x16x4_f32(
                   false, a, false, bf0, (short)0, acc0, false, false);
        acc1 = __builtin_amdgcn_wmma_f32_16x16x4_f32(
                   false, a, false, bf1, (short)0, acc1, false, false);
    }

    // ---- transpose via LDS; then b128 coalesced stores along ij ----------
    #pragma unroll
    for (int rr = 0; rr < 8; ++rr) {
        const int ml = wm * 16 + rr + (kh ? 8 : 0);   // local M of acc[rr]
        T[(wn * 32 + l15) * TSTR + ml]      = acc0[rr];
        T[(wn * 32 + 16 + l15) * TSTR + ml] = acc1[rr];
    }
    __syncthreads();

    for (int q = tid; q < COUT * 16; q += 256) {       // 4 iterations
        const int n  = q >> 4;                         // output channel
        const int mq = (q & 15) << 2;                  // 4 consecutive pixels
        const float4 vv = *(const float4*)&T[n * TSTR + mq];   // ds_load_b128
        *(float4*)&out[((size_t)bb * COUT + n) * NPTS + ij0 + mq] = vv;
    }
}

// ---------------------------------------------------------------------------
extern "C" void kernel_launch(void* const* d_in, const int* in_sizes, int n_in,
                              void* d_out, int out_size, void* d_ws, size_t ws_size,
                              hipStream_t stream)
{
    const float* inputs = (const float*)d_in[0];   // [64,3,224,224] f32
    const float* weight = (const float*)d_in[1];   // [64,3,7,7] f32
    float* out = (float*)d_out;                    // [64,64,112,112] f32
    int* geo = (int*)d_ws;                         // 3 * NPTS ints

    lgn_geometry_kernel<<<1, 256, 0, stream>>>(geo);

    const int M_total = 64 * NPTS;                 // 802816
    const int grid    = M_total / MBLK;            // 12544
    lgn_wmma_kernel<<<grid, 256, 0, stream>>>(inputs, weight, geo, out);
}